// SDLLoss_8366596292727
// MI455X (gfx1250) — compile-verified
//
#include <hip/hip_runtime.h>
#include <hip/hip_bf16.h>
#include <math.h>
#include <float.h>

#define EMBED  300
#define KDIM   49
#define BATCH  128
#define CDIM   1000
#define WEIGHTC 0.3f

#define NTILE  63                 // ceil(1000/16)
#define TASKS  (BATCH * NTILE)    // 8064 wave-tasks == 1008 blocks * 8 waves exactly

typedef float v2f __attribute__((ext_vector_type(2)));
typedef float v8f __attribute__((ext_vector_type(8)));

// ---------------------------------------------------------------------------
// Kernel 1: scores[b] = x[b]^T (49x300) @ wordvec (300x1000)  via
//           V_WMMA_F32_16X16X4_F32, then dot[b,c] = max_k scores[b,k,c].
// One wave32 per (b, 16-wide C tile). M (the 49 k-rows) = 4 tiles of 16.
// Out-of-range columns/rows are handled by ADDRESS CLAMPING (the garbage
// lands only in D entries that are never consumed), so the K-loop is
// completely branch-free: one load clause + 4 WMMAs per iteration.
// ---------------------------------------------------------------------------
__global__ __launch_bounds__(256) void k_gemm_max(const float* __restrict__ x,
                                                  const float* __restrict__ wv,
                                                  float* __restrict__ dot)
{
    const int wave = (blockIdx.x * blockDim.x + threadIdx.x) >> 5;
    const int lane = threadIdx.x & 31;
    if (wave >= TASKS) return;           // never taken (exact fit) -> EXEC stays all-1s

    const int b  = wave / NTILE;
    const int n0 = (wave % NTILE) * 16;
    const int ln = lane & 15;            // M (for A) / N (for B) position
    const int hi = lane >> 4;            // selects K-pair {0,1} vs {2,3}
    const int n  = n0 + ln;
    const int nc = (n < CDIM) ? n : (CDIM - 1);   // clamped column for loads

    v8f acc0 = {}, acc1 = {}, acc2 = {}, acc3 = {};

    const float* xb   = x  + (size_t)b * (EMBED * KDIM);
    const float* wcol = wv + nc;         // column-n pointer into wordvec

    for (int e = 0; e < EMBED; e += 4) {
        const int e0 = e + 2 * hi;       // this half-wave's K indices: e0, e0+1
        const int e1 = e0 + 1;

        // B fragment: 4x16 slice of wordvec, N = n (clamped), rows K = e0,e1
        v2f bf;
        bf.x = wcol[e0 * CDIM];
        bf.y = wcol[e1 * CDIM];

        // speculative prefetch of next wordvec panel (global_prefetch_b8);
        // OOB prefetches are silently dropped by hardware.
        __builtin_prefetch(&wcol[(e0 + 4) * CDIM]);

        // A fragments: rows m of x[b] (m = k-index), cols K = e0,e1.
        // Tile 3 broadcasts row 48 to all lanes: rows 49..63 of D are garbage
        // but excluded from the max below.
        const float* r0 = xb + e0 * KDIM;
        const float* r1 = xb + e1 * KDIM;
        v2f a0, a1, a2, a3;
        a0.x = r0[ln];        a0.y = r1[ln];         // m = 0..15
        a1.x = r0[16 + ln];   a1.y = r1[16 + ln];    // m = 16..31
        a2.x = r0[32 + ln];   a2.y = r1[32 + ln];    // m = 32..47
        a3.x = r0[48];        a3.y = r1[48];         // m = 48 (broadcast)

        // 4 back-to-back identical WMMAs sharing bf -> reuse_b hint on 2..4
        acc0 = __builtin_amdgcn_wmma_f32_16x16x4_f32(false, a0, false, bf, (short)0, acc0, false, false);
        acc1 = __builtin_amdgcn_wmma_f32_16x16x4_f32(false, a1, false, bf, (short)0, acc1, false, true);
        acc2 = __builtin_amdgcn_wmma_f32_16x16x4_f32(false, a2, false, bf, (short)0, acc2, false, true);
        acc3 = __builtin_amdgcn_wmma_f32_16x16x4_f32(false, a3, false, bf, (short)0, acc3, false, true);
    }

    // Per-lane max over the rows this lane holds (C/D layout: VGPR r -> M = r + 8*hi).
    // Tiles 0..2 cover m=0..47 (all valid rows).
    float mx = -FLT_MAX;
#pragma unroll
    for (int r = 0; r < 8; ++r)
        mx = fmaxf(mx, fmaxf(acc0[r], fmaxf(acc1[r], acc2[r])));
    // Tile 3: only row m=48 is real = VGPR 0 on the hi==0 half-wave.
    if (hi == 0) mx = fmaxf(mx, acc3[0]);

    // Combine the two half-waves holding the same column n
    float other = __shfl_xor(mx, 16, 32);
    mx = fmaxf(mx, other);

    if (hi == 0 && n < CDIM) dot[b * CDIM + n] = mx;
}

// ---------------------------------------------------------------------------
// Block-wide sum reduction (blockDim.x == 256)
// ---------------------------------------------------------------------------
__device__ __forceinline__ float blockSum(float v, float* red)
{
    const int t = threadIdx.x;
    red[t] = v; __syncthreads();
#pragma unroll
    for (int s = 128; s > 0; s >>= 1) {
        if (t < s) red[t] += red[t + s];
        __syncthreads();
    }
    float r = red[0]; __syncthreads();
    return r;
}

// ---------------------------------------------------------------------------
// Kernel 2: one block per batch b -> per-batch loss l[b]
// ---------------------------------------------------------------------------
__global__ __launch_bounds__(256) void k_loss(const float* __restrict__ x,
                                              const float* __restrict__ wv,
                                              const int*   __restrict__ y,
                                              const float* __restrict__ dot,
                                              float* __restrict__ lvec)
{
    __shared__ float sdot[CDIM];
    __shared__ float spos[CDIM];
    __shared__ float red[256];

    const int b = blockIdx.x;
    const int t = threadIdx.x;

    for (int c = t; c < CDIM; c += 256) {
        sdot[c] = dot[b * CDIM + c];
        spos[c] = (y[b * CDIM + c] == 1) ? 1.f : 0.f;
    }
    __syncthreads();

    // num_pos / num_neg
    float npl = 0.f;
    for (int c = t; c < CDIM; c += 256) npl += spos[c];
    const float num_pos = blockSum(npl, red);
    const float num_neg = (float)CDIM - num_pos;

    // pair_sum = sum_{i pos, j neg} log1p(exp(dot_j - dot_i))
    float ps = 0.f;
    for (int p = t; p < CDIM * CDIM; p += 256) {
        const int i = p / CDIM;
        const int j = p - i * CDIM;
        if (spos[i] > 0.f && spos[j] == 0.f)
            ps += log1pf(expf(sdot[j] - sdot[i]));
    }
    const float pair_sum = blockSum(ps, red);

    // alt = sum_pos log1p(exp(-dot))
    float al = 0.f;
    for (int c = t; c < CDIM; c += 256)
        al += spos[c] * log1pf(expf(-sdot[c]));
    const float alt = blockSum(al, red);

    // total_var: per-e variance of wordvec over pos columns
    float tv = 0.f;
    for (int e = t; e < EMBED; e += 256) {
        float s1 = 0.f, s2 = 0.f;
        const float* wr = wv + (size_t)e * CDIM;
        for (int c = 0; c < CDIM; ++c) {
            const float w = wr[c];
            const float p = spos[c];
            s1 += w * p;
            s2 += w * w * p;
        }
        const float npm = fmaxf(num_pos, 1.f);
        tv += (s2 - s1 * s1 / npm) / fmaxf(num_pos - 1.f, 1.f);
    }
    float total_var = blockSum(tv, red);
    if (!(num_pos > 1.f)) total_var = 0.f;

    // l1_err = sum_e |var_k(x[b,e,:])| (ddof=1)
    float le = 0.f;
    for (int e = t; e < EMBED; e += 256) {
        const float* row = x + ((size_t)b * EMBED + e) * KDIM;
        float s = 0.f, ss = 0.f;
        for (int k = 0; k < KDIM; ++k) {
            const float v = row[k];
            s += v; ss += v * v;
        }
        const float var = (ss - s * s / (float)KDIM) / (float)(KDIM - 1);
        le += fabsf(var);
    }
    const float l1 = blockSum(le, red);

    if (t == 0) {
        const float base = (num_neg > 0.f) ? pair_sum : alt;
        float l = (1.f + total_var) * base / num_pos;
        l = 2.f * ((1.f - WEIGHTC) * l + WEIGHTC * l1);
        lvec[b] = l;
    }
}

// ---------------------------------------------------------------------------
// Kernel 3: mean over the 128 batch losses
// ---------------------------------------------------------------------------
__global__ void k_mean(const float* __restrict__ lvec, float* __restrict__ out)
{
    __shared__ float red[BATCH];
    const int t = threadIdx.x;
    red[t] = lvec[t];
    __syncthreads();
#pragma unroll
    for (int s = BATCH / 2; s > 0; s >>= 1) {
        if (t < s) red[t] += red[t + s];
        __syncthreads();
    }
    if (t == 0) out[0] = red[0] / (float)BATCH;
}

extern "C" void kernel_launch(void* const* d_in, const int* in_sizes, int n_in,
                              void* d_out, int out_size, void* d_ws, size_t ws_size,
                              hipStream_t stream)
{
    const float* x  = (const float*)d_in[0];   // (128, 300*49)
    const float* wv = (const float*)d_in[1];   // (300, 1000)
    const int*   y  = (const int*)d_in[2];     // (128, 1000)

    float* dot  = (float*)d_ws;                // 128*1000 floats
    float* lvec = dot + BATCH * CDIM;          // 128 floats

    const int blocks1 = (TASKS + 7) / 8;       // 8 waves (256 threads) per block
    k_gemm_max<<<blocks1, 256, 0, stream>>>(x, wv, dot);
    k_loss<<<BATCH, 256, 0, stream>>>(x, wv, y, dot, lvec);
    k_mean<<<1, BATCH, 0, stream>>>(lvec, (float*)d_out);
}